// NTXentLoss_84490596647265
// MI455X (gfx1250) — compile-verified
//
#include <hip/hip_runtime.h>

typedef __attribute__((ext_vector_type(2))) float v2f;
typedef __attribute__((ext_vector_type(8))) float v8f;

// Exact parameter types for the async global->LDS builtin (confirmed by hipcc
// diagnostic: param 0 is 'int4 __device__*', i.e. non-const int4 in AS1).
typedef __attribute__((__vector_size__(4 * sizeof(int)))) int v4i;
typedef __attribute__((address_space(1))) v4i* gv4i_p;   // global (__device__)
typedef __attribute__((address_space(3))) v4i* sv4i_p;   // LDS (__shared__)

static constexpr int ZD     = 256;    // feature dim
static constexpr int HALFN  = 4096;   // N
static constexpr int NROWS  = 8192;   // 2N
static constexpr int NSEG   = 8;      // column segments (512 tiles / 64)
static constexpr int SEGT   = 64;     // tiles per segment
static constexpr float INVT = 2.0f;   // 1 / TEMPERATURE
// logits expressed in base-2 units: val = dot * (1/T) * log2(e)
static constexpr float SCALE2 = 2.885390081777927f;   // 2 * log2(e)
static constexpr float LN2    = 0.6931471805599453f;
static constexpr float NEG_BIG = -3.0e38f;

#if __has_builtin(__builtin_amdgcn_global_load_async_to_lds_b128)
#define USE_ASYNC_LDS 1
#else
#define USE_ASYNC_LDS 0
#endif

__device__ __forceinline__ void wait_asynccnt0() {
#if __has_builtin(__builtin_amdgcn_s_wait_asynccnt)
  __builtin_amdgcn_s_wait_asynccnt(0);
#else
  asm volatile("s_wait_asynccnt 0x0" ::: "memory");
#endif
}

__device__ __forceinline__ const float* row_ptr(const float* zi, const float* zj, int r) {
  return (r < HALFN) ? (zi + (size_t)r * ZD) : (zj + (size_t)(r - HALFN) * ZD);
}

__device__ __forceinline__ const float* tile_ptr(const float* zi, const float* zj, int cb) {
  return (cb < 256) ? (zi + (size_t)cb * 16 * ZD) : (zj + (size_t)(cb - 256) * 16 * ZD);
}

// Kernel 1: streaming fp32 WMMA GEMM + online logsumexp (partial per column segment).
// Grid: 256 workgroups = 8 column segments x 32 row groups; 256 threads (8 waves).
// WG = 256 rows x 1024 columns; wave = 32 rows (two resident 16-row A panels).
__global__ __launch_bounds__(256) void ntxent_lse_kernel(
    const float* __restrict__ zi, const float* __restrict__ zj,
    float* __restrict__ partM, float* __restrict__ partS) {
  // Double-buffered B tile: 16 columns (z rows) x 256 K, rows padded to 260 floats
  __shared__ __align__(16) float Blds[2][16 * 260];

  const int tid  = threadIdx.x;
  const int wave = tid >> 5;
  const int lane = tid & 31;
  const int half = lane >> 4;   // K-pair select for A/B, row-group select for C
  const int nl   = lane & 15;   // M for A, N for B/C

  const int rg   = blockIdx.x & 31;   // row group
  const int seg  = blockIdx.x >> 5;   // column segment
  const int cbeg = seg * SEGT;        // first column tile of this segment

  const int rbase = rg * 256 + wave * 32;  // this wave's 32-row strip

  // Two resident A panels. A(16x4 f32): lane holds M=nl, K = ks*4 + half*2 + {0,1}
  const float* arow0 = row_ptr(zi, zj, rbase + nl);
  const float* arow1 = row_ptr(zi, zj, rbase + 16 + nl);
  v2f regA0[64], regA1[64];
  #pragma unroll
  for (int ks = 0; ks < 64; ++ks) {
    regA0[ks] = *(const v2f*)(arow0 + ks * 4 + half * 2);
    regA1[ks] = *(const v2f*)(arow1 + ks * 4 + half * 2);
  }

  float m0[8], s0[8], m1[8], s1[8];
  #pragma unroll
  for (int v = 0; v < 8; ++v) { m0[v] = NEG_BIG; s0[v] = 0.f; m1[v] = NEG_BIG; s1[v] = 0.f; }

  const int dt0 = rbase >> 4;       // diagonal tiles for the two panels (global tile idx)
  const int dt1 = dt0 + 1;

#if USE_ASYNC_LDS
  // Async global->LDS copy of one contiguous 16KB tile; no VGPR staging, no spills.
  auto issue_tile = [&](int cb, int buf) {
    const char* gsrc = (const char*)tile_ptr(zi, zj, cb);
    #pragma unroll
    for (int i = 0; i < 4; ++i) {
      int idx4 = tid + i * 256;     // 1024 x 16B chunks
      int row  = idx4 >> 6;
      int k4   = idx4 & 63;
      __builtin_amdgcn_global_load_async_to_lds_b128(
          (gv4i_p)(gsrc + (size_t)idx4 * 16),
          (sv4i_p)(&Blds[buf][row * 260 + k4 * 4]),
          0, 0);
    }
  };

  issue_tile(cbeg, 0);
  wait_asynccnt0();
  __syncthreads();
#else
  {
    const float4* src = (const float4*)tile_ptr(zi, zj, cbeg);
    #pragma unroll
    for (int i = 0; i < 4; ++i) {
      int idx4 = tid + i * 256;
      *(float4*)(&Blds[0][(idx4 >> 6) * 260 + (idx4 & 63) * 4]) = src[idx4];
    }
  }
  __syncthreads();
#endif

  for (int t = 0; t < SEGT; ++t) {
    const int cb = cbeg + t;
#if USE_ASYNC_LDS
    if (t + 1 < SEGT) issue_tile(cb + 1, (t + 1) & 1);
#endif

    // Two 16x16 sim tiles: 64 k-steps, 2 WMMA per B pair-load
    const float* bcol = &Blds[t & 1][nl * 260 + half * 2];
    v8f c0 = {0.f,0.f,0.f,0.f,0.f,0.f,0.f,0.f};
    v8f c1 = {0.f,0.f,0.f,0.f,0.f,0.f,0.f,0.f};
    #pragma unroll
    for (int ks = 0; ks < 64; ++ks) {
      v2f b = *(const v2f*)(bcol + ks * 4);
      c0 = __builtin_amdgcn_wmma_f32_16x16x4_f32(false, regA0[ks], false, b,
                                                 (short)0, c0, false, false);
      c1 = __builtin_amdgcn_wmma_f32_16x16x4_f32(false, regA1[ks], false, b,
                                                 (short)0, c1, false, false);
    }

    // Online per-lane logsumexp (base-2). Diagonal masking only in the (wave-
    // uniform) iterations whose column tile intersects this wave's rows.
    if (cb == dt0 || cb == dt1) {
      const int col = cb * 16 + nl;
      #pragma unroll
      for (int v = 0; v < 8; ++v) {
        float val = c0[v] * SCALE2;
        int row = rbase + half * 8 + v;                  // C layout: M = v + half*8
        if (col == row) val = NEG_BIG;                   // mask self-similarity
        float mn = (m0[v] > val) ? m0[v] : val;
        s0[v] = s0[v] * exp2f(m0[v] - mn) + exp2f(val - mn);
        m0[v] = mn;
      }
      #pragma unroll
      for (int v = 0; v < 8; ++v) {
        float val = c1[v] * SCALE2;
        int row = rbase + 16 + half * 8 + v;
        if (col == row) val = NEG_BIG;
        float mn = (m1[v] > val) ? m1[v] : val;
        s1[v] = s1[v] * exp2f(m1[v] - mn) + exp2f(val - mn);
        m1[v] = mn;
      }
    } else {
      #pragma unroll
      for (int v = 0; v < 8; ++v) {
        float val = c0[v] * SCALE2;
        float mn = (m0[v] > val) ? m0[v] : val;
        s0[v] = s0[v] * exp2f(m0[v] - mn) + exp2f(val - mn);
        m0[v] = mn;
      }
      #pragma unroll
      for (int v = 0; v < 8; ++v) {
        float val = c1[v] * SCALE2;
        float mn = (m1[v] > val) ? m1[v] : val;
        s1[v] = s1[v] * exp2f(m1[v] - mn) + exp2f(val - mn);
        m1[v] = mn;
      }
    }

#if USE_ASYNC_LDS
    wait_asynccnt0();     // this wave's async writes for tile t+1 are in LDS
    __syncthreads();      // all waves done (compute on cur + async into nxt)
#else
    __syncthreads();
    if (t + 1 < SEGT) {
      const float4* src = (const float4*)tile_ptr(zi, zj, cb + 1);
      #pragma unroll
      for (int i = 0; i < 4; ++i) {
        int idx4 = tid + i * 256;
        *(float4*)(&Blds[(t + 1) & 1][(idx4 >> 6) * 260 + (idx4 & 63) * 4]) = src[idx4];
      }
    }
    __syncthreads();
#endif
  }

  // Merge the 16 column-owning lanes within each half of the wave
  #pragma unroll
  for (int xm = 1; xm <= 8; xm <<= 1) {
    #pragma unroll
    for (int v = 0; v < 8; ++v) {
      float m2 = __shfl_xor(m0[v], xm, 32);
      float sx = __shfl_xor(s0[v], xm, 32);
      float mn = (m0[v] > m2) ? m0[v] : m2;
      s0[v] = s0[v] * exp2f(m0[v] - mn) + sx * exp2f(m2 - mn);
      m0[v] = mn;

      m2 = __shfl_xor(m1[v], xm, 32);
      sx = __shfl_xor(s1[v], xm, 32);
      mn = (m1[v] > m2) ? m1[v] : m2;
      s1[v] = s1[v] * exp2f(m1[v] - mn) + sx * exp2f(m2 - mn);
      m1[v] = mn;
    }
  }

  if (nl == 0) {
    const int base = seg * NROWS + rbase + half * 8;
    #pragma unroll
    for (int v = 0; v < 8; ++v) {
      partM[base + v]      = m0[v];
      partS[base + v]      = s0[v];
      partM[base + 16 + v] = m1[v];
      partS[base + 16 + v] = s1[v];
    }
  }
}

// Kernel 1b: merge the NSEG partial (m, s) pairs per row -> lse (natural log units).
__global__ __launch_bounds__(256) void ntxent_merge_kernel(
    const float* __restrict__ partM, const float* __restrict__ partS,
    float* __restrict__ lse) {
  const int r = blockIdx.x * 256 + threadIdx.x;
  float m = NEG_BIG, s = 0.f;
  #pragma unroll
  for (int g = 0; g < NSEG; ++g) {
    float mg = partM[g * NROWS + r];
    float sg = partS[g * NROWS + r];
    float mn = (m > mg) ? m : mg;
    s = s * exp2f(m - mn) + sg * exp2f(mg - mn);
    m = mn;
  }
  lse[r] = LN2 * (m + log2f(s));
}

// Kernel 2: positive-pair logits, exact fp32. One wave per row.
__global__ __launch_bounds__(32) void ntxent_pos_kernel(
    const float* __restrict__ zi, const float* __restrict__ zj,
    float* __restrict__ pos) {
  const int r   = blockIdx.x;
  const int lab = (r < HALFN) ? (r + HALFN) : (r - HALFN);
  const float4* a = (const float4*)row_ptr(zi, zj, r);
  const float4* b = (const float4*)row_ptr(zi, zj, lab);
  const int lane = threadIdx.x;
  float acc = 0.f;
  #pragma unroll
  for (int i = 0; i < 2; ++i) {
    float4 av = a[lane * 2 + i];
    float4 bv = b[lane * 2 + i];
    acc += av.x * bv.x + av.y * bv.y + av.z * bv.z + av.w * bv.w;
  }
  #pragma unroll
  for (int xm = 16; xm >= 1; xm >>= 1) acc += __shfl_xor(acc, xm, 32);
  if (lane == 0) pos[r] = acc * INVT;
}

// Kernel 3: deterministic single-block final reduction.
__global__ __launch_bounds__(256) void ntxent_reduce_kernel(
    const float* __restrict__ lse, const float* __restrict__ pos,
    float* __restrict__ out) {
  __shared__ float red[256];
  const int tid = threadIdx.x;
  float acc = 0.f;
  for (int r = tid; r < NROWS; r += 256) acc += lse[r] - pos[r];
  red[tid] = acc;
  __syncthreads();
  for (int s = 128; s > 0; s >>= 1) {
    if (tid < s) red[tid] += red[tid + s];
    __syncthreads();
  }
  if (tid == 0) out[0] = red[0] / (float)NROWS;
}

extern "C" void kernel_launch(void* const* d_in, const int* in_sizes, int n_in,
                              void* d_out, int out_size, void* d_ws, size_t ws_size,
                              hipStream_t stream) {
  (void)in_sizes; (void)n_in; (void)out_size; (void)ws_size;
  const float* zi = (const float*)d_in[0];
  const float* zj = (const float*)d_in[1];
  float* partM = (float*)d_ws;                 // NSEG * NROWS floats
  float* partS = partM + NSEG * NROWS;         // NSEG * NROWS floats
  float* lse   = partS + NSEG * NROWS;         // NROWS floats
  float* pos   = lse + NROWS;                  // NROWS floats

  ntxent_lse_kernel<<<NSEG * 32, 256, 0, stream>>>(zi, zj, partM, partS);
  ntxent_merge_kernel<<<NROWS / 256, 256, 0, stream>>>(partM, partS, lse);
  ntxent_pos_kernel<<<NROWS, 32, 0, stream>>>(zi, zj, pos);
  ntxent_reduce_kernel<<<1, 256, 0, stream>>>(lse, pos, (float*)d_out);
}